// Attention_81028853007030
// MI455X (gfx1250) — compile-verified
//
#include <hip/hip_runtime.h>
#include <math.h>

// Problem constants (from reference): B=64, L=2048, E=300, H=1 -> 4H=4 gates.
#define BB 64
#define LL 2048
#define EE 300
#define GG 4

typedef float v2f __attribute__((ext_vector_type(2)));
typedef float v8f __attribute__((ext_vector_type(8)));

// gfx1250 async global->LDS path (ASYNCcnt), guarded so compile never breaks.
#if defined(__AMDGCN__) && __has_builtin(__builtin_amdgcn_global_load_async_to_lds_b128) && __has_builtin(__builtin_amdgcn_s_wait_asynccnt)
#define USE_ASYNC_LDS 1
#else
#define USE_ASYNC_LDS 0
#endif

// Exact parameter types per hipcc diagnostic:
//   arg0: __attribute__((vector_size(16))) int  AS1*   (global)
//   arg1: __attribute__((vector_size(16))) int  AS3*   (LDS)
typedef int v4i __attribute__((vector_size(16)));
typedef __attribute__((address_space(1))) v4i* gv4i_p;
typedef __attribute__((address_space(3))) v4i* lv4i_p;

// ---------------------------------------------------------------------------
// Phase 1: xg[row][g] = sum_e x[row][e] * W_ih[g][e]   (row = b*L + t)
// V_WMMA_F32_16X16X4_F32: A = 16x4 f32 tile of x, B = 4x16 tile of W_ih^T
// (stored as a 16x300 LDS tile with rows 4..15 pre-zeroed -> no inner-loop
// masking), C/D = 16x16 f32 accumulator; 75 WMMAs cover K=300.
//
// Block = 64 threads (2 waves), 32 contiguous rows of x per block.
// Stride-300 LDS layout: column access banks are 44*m mod 64 (16 distinct
// residues =0 mod 4) for lanes 0-15 and the same +2 for lanes 16-31 -> the
// wave's b64 reads touch all 64 banks, conflict-free, and the staging copy is
// a flat 16B-aligned chunk (1200 B/row) ideal for async b128 loads to LDS.
// ---------------------------------------------------------------------------
#define RPB 32

__global__ __launch_bounds__(64) void xg_gemm_wmma(const float* __restrict__ x,
                                                   const float* __restrict__ W_ih,
                                                   float* __restrict__ xg) {
  __shared__ float sA[RPB * EE];   // 38,400 B : 32 rows of x
  __shared__ float sW[16 * EE];    // 19,200 B : W_ih padded to 16 rows (4..15 = 0)

  const int tid = threadIdx.x;
  const long rowBase = (long)blockIdx.x * RPB;

  const float4* src4 = (const float4*)(x + rowBase * EE);  // 16B aligned (1200 B rows)
  float4*       sA4  = (float4*)sA;
  const float4* w4   = (const float4*)W_ih;
  float4*       sW4  = (float4*)sW;

  const float4 z4 = make_float4(0.f, 0.f, 0.f, 0.f);

#if USE_ASYNC_LDS
  // Async DMA of the x tile (2400 float4) and W_ih (300 float4) into LDS.
  for (int i = tid; i < (RPB * EE) / 4; i += 64)
    __builtin_amdgcn_global_load_async_to_lds_b128((gv4i_p)(src4 + i),
                                                   (lv4i_p)(sA4 + i), 0, 0);
  for (int i = tid; i < (GG * EE) / 4; i += 64)
    __builtin_amdgcn_global_load_async_to_lds_b128((gv4i_p)(w4 + i),
                                                   (lv4i_p)(sW4 + i), 0, 0);
  // Zero the padded B rows (4..15) while the DMA is in flight.
  for (int i = tid + (GG * EE) / 4; i < (16 * EE) / 4; i += 64) sW4[i] = z4;
  __builtin_amdgcn_s_wait_asynccnt(0);
#else
  for (int i = tid; i < (RPB * EE) / 4; i += 64) sA4[i] = src4[i];
  for (int i = tid; i < (GG * EE) / 4; i += 64) sW4[i] = w4[i];
  for (int i = tid + (GG * EE) / 4; i < (16 * EE) / 4; i += 64) sW4[i] = z4;
#endif
  __syncthreads();

  const int wave = tid >> 5;       // 0..1 : which 16-row tile
  const int lane = tid & 31;
  const int half = lane >> 4;      // 0..1 : K sub-pair select
  const int mn   = lane & 15;      // A: row M | B/D: col N
  const int tileRow = wave * 16;

  // ISA VGPR layouts (CDNA5 ISA):
  //   A 16x4 f32 : v0 = A[m][2*half],  v1 = A[m][2*half+1]
  //   B 4x16 f32 : v0 = B[2*half][n],  v1 = B[2*half+1][n],  B[k][n] = W_ih[n][k]
  //   D 16x16 f32: acc[r] = D[r + 8*half][n]
  const float* aRow = &sA[(tileRow + mn) * EE];
  const float* bRow = &sW[mn * EE];          // rows >= 4 are zero -> no masking

  v8f acc = {0.f, 0.f, 0.f, 0.f, 0.f, 0.f, 0.f, 0.f};

  for (int k0 = 0; k0 < EE; k0 += 4) {       // 75 WMMA steps
    const int k = k0 + 2 * half;
    v2f a; a.x = aRow[k]; a.y = aRow[k + 1];
    v2f b; b.x = bRow[k]; b.y = bRow[k + 1];
    acc = __builtin_amdgcn_wmma_f32_16x16x4_f32(
        /*neg_a=*/false, a, /*neg_b=*/false, b,
        /*c_mod=*/(short)0, acc, /*reuse_a=*/false, /*reuse_b=*/false);
  }

  // Scatter the 4 valid gate columns of the 16x16 result.
  if (mn < GG) {
    #pragma unroll
    for (int r = 0; r < 8; ++r) {
      long row = rowBase + tileRow + r + 8 * half;
      xg[row * GG + mn] = acc[r];
    }
  }
}

// ---------------------------------------------------------------------------
// Phase 2: sequential LSTM over L (nonlinear recurrence -> cannot parallelize
// over time). One thread per batch row (64 threads = 2 waves). Fuses the
// (faithful, buggy) t==0 mask, online softmax, then converts scores ->
// attention weights in place.
// ---------------------------------------------------------------------------
__global__ __launch_bounds__(64) void lstm_scan(const float* __restrict__ xg,
                                                const int*   __restrict__ lens,
                                                const float* __restrict__ W_hh,
                                                const float* __restrict__ b_ih,
                                                const float* __restrict__ b_hh,
                                                float* __restrict__ attn) {
  const int b = threadIdx.x;     // 0..63
  const float w0 = W_hh[0], w1 = W_hh[1], w2 = W_hh[2], w3 = W_hh[3];
  const float bi = b_ih[0] + b_hh[0];
  const float bf = b_ih[1] + b_hh[1];
  const float bg = b_ih[2] + b_hh[2];
  const float bo = b_ih[3] + b_hh[3];
  const int len = lens[b];

  const float4* gp = ((const float4*)xg) + (long)b * LL;  // 4 gates = 1 float4/step
  float* arow = attn + (long)b * LL;

  float h = 0.f, c = 0.f;
  float mx = -__builtin_inff();
  float s = 0.f;

  for (int t = 0; t < LL; ++t) {
    float4 g = gp[t];
    float gi = g.x + bi + h * w0;
    float gf = g.y + bf + h * w1;
    float gc = g.z + bg + h * w2;
    float go = g.w + bo + h * w3;
    float ii = 1.0f / (1.0f + __expf(-gi));
    float ff = 1.0f / (1.0f + __expf(-gf));
    float gg = tanhf(gc);
    float oo = 1.0f / (1.0f + __expf(-go));
    c = ff * c + ii * gg;
    h = oo * tanhf(c);

    float sc = (t == 0 && len > 0) ? -__builtin_inff() : h;
    arow[t] = sc;
    if (sc != -__builtin_inff()) {          // masked entry contributes 0
      float nm = fmaxf(mx, sc);
      s = s * __expf(mx - nm) + __expf(sc - nm);
      mx = nm;
    }
  }

  const float inv = 1.0f / s;
  for (int t = 0; t < LL; ++t) {
    float sc = arow[t];
    arow[t] = (sc == -__builtin_inff()) ? 0.0f : __expf(sc - mx) * inv;
  }
}

// ---------------------------------------------------------------------------
// Phase 3: out[b][t][e] = attn[b][t] * x[b][t][e].
// Rows are 300 floats = 1200 B (16B aligned) -> pure float4 streaming; this
// phase + phase 1 set the memory-bound floor (~470 MB total traffic ~ 20 us).
// ---------------------------------------------------------------------------
__global__ __launch_bounds__(256) void scale_out(const float* __restrict__ x,
                                                 const float* __restrict__ attn,
                                                 float* __restrict__ out,
                                                 unsigned n4) {
  unsigned idx = blockIdx.x * 256u + threadIdx.x;
  if (idx >= n4) return;
  unsigned row = idx / (EE / 4);                 // 75 float4 per row
  float w = attn[row];
  float4 v = ((const float4*)x)[idx];
  v.x *= w; v.y *= w; v.z *= w; v.w *= w;
  ((float4*)out)[idx] = v;
}

// ---------------------------------------------------------------------------
extern "C" void kernel_launch(void* const* d_in, const int* in_sizes, int n_in,
                              void* d_out, int out_size, void* d_ws, size_t ws_size,
                              hipStream_t stream) {
  const float* x     = (const float*)d_in[0];   // [B, L, E]
  const int*   lens  = (const int*)  d_in[1];   // [B]
  const float* W_ih  = (const float*)d_in[2];   // [4, E]
  const float* W_hh  = (const float*)d_in[3];   // [4, 1]
  const float* b_ih  = (const float*)d_in[4];   // [4]
  const float* b_hh  = (const float*)d_in[5];   // [4]
  float* out = (float*)d_out;                   // [B, L, E]

  float* xg   = (float*)d_ws;                          // B*L*4 floats (2 MB)
  float* attn = xg + (size_t)BB * LL * GG;             // B*L floats  (0.5 MB)

  xg_gemm_wmma<<<(BB * LL) / RPB, 64, 0, stream>>>(x, W_ih, xg);
  lstm_scan<<<1, BB, 0, stream>>>(xg, lens, W_hh, b_ih, b_hh, attn);

  const unsigned n4 = BB * LL * (EE / 4);              // 9,830,400 float4s
  scale_out<<<(n4 + 255u) / 256u, 256, 0, stream>>>(x, attn, out, n4);
}